// GINNodeClassifier_1133871366241
// MI455X (gfx1250) — compile-verified
//
#include <hip/hip_runtime.h>

typedef __attribute__((ext_vector_type(2))) float v2f;
typedef __attribute__((ext_vector_type(8))) float v8f;

#define IN_DIM   128
#define H_DIM    256
#define OUT_DIM  64
#define M_TILE   64
#define LDS_STR  260   // 256 + 4 floats pad; 260*4B row stride => 16B aligned, bank-conflict-free cols

// ---------------------------------------------------------------- zero
__global__ void zero_f32(float* __restrict__ p, long long count) {
  long long i = (long long)blockIdx.x * blockDim.x + threadIdx.x;
  long long stride = (long long)gridDim.x * blockDim.x;
  for (; i < count; i += stride) p[i] = 0.0f;
}

// ---------------------------------------------------------------- weight transpose  out[c][r] = in[r][c]
// Tiny (<=256KB) matrices; gives contiguous K-pairs for WMMA B fragments (b64 loads).
__global__ void transpose_f32(const float* __restrict__ in, float* __restrict__ out,
                              int R, int C) {
  int i = blockIdx.x * blockDim.x + threadIdx.x;
  if (i >= R * C) return;
  int r = i / C, c = i - r * C;
  out[c * R + r] = in[i];
}

// ---------------------------------------------------------------- scatter-add aggregation
// agg[dst[e]] += x[src[e]]  (d floats/edge). agg (51MB) is L2-resident (192MB L2) -> atomics
// resolve in L2 without HBM round trips.
__global__ void scatter_add_f32(const float* __restrict__ x,
                                const int* __restrict__ src,
                                const int* __restrict__ dst,
                                float* __restrict__ agg,
                                int E, int d) {
  const int dq = d >> 2;
  long long idx = (long long)blockIdx.x * blockDim.x + threadIdx.x;
  long long total = (long long)E * dq;
  if (idx >= total) return;
  int e = (int)(idx / dq);
  int f = ((int)(idx % dq)) << 2;
  int s = src[e];
  int t = dst[e];
  const float4 v = *(const float4*)(x + (long long)s * d + f);
  float* p = agg + (long long)t * d + f;
  atomicAdd(p + 0, v.x);
  atomicAdd(p + 1, v.y);
  atomicAdd(p + 2, v.z);
  atomicAdd(p + 3, v.w);
}

// ---------------------------------------------------------------- fused GIN MLP layer
// h = (1+eps)*X + AGG ; Y = relu(h@W1 + b1) ; Z = Y@W2 + b2 ; optional BN + leaky-relu.
// 256 threads (8 wave32) per workgroup, 64-row tile, fp32 WMMA 16x16x4.
// WT1 = W1^T  [H_DIM][DIN],  WT2 = W2^T  [DOUT][H_DIM]  (K contiguous -> b64 B-fragment loads).
// Compile-time DIN/DOUT -> exact unrolls, no remainder loops.
template <int DIN, int DOUT, int DO_BN>
__global__ __launch_bounds__(256)
void gin_mlp(const float* __restrict__ X,
             const float* __restrict__ AGG,
             const float* __restrict__ epsp,       // nullptr -> eps = 0
             const float* __restrict__ WT1,
             const float* __restrict__ B1,
             const float* __restrict__ WT2,
             const float* __restrict__ B2,
             const float* __restrict__ bn_gamma,   // nullptr when DO_BN==0
             const float* __restrict__ bn_beta,
             const float* __restrict__ bn_mean,
             const float* __restrict__ bn_var,
             float* __restrict__ OUT,              // [n, DOUT]
             int n) {
  __shared__ float sH[M_TILE][LDS_STR];
  __shared__ float sY[M_TILE][LDS_STR];

  const int tid  = threadIdx.x;
  const int wave = tid >> 5;
  const int lane = tid & 31;
  const int row0 = blockIdx.x * M_TILE;
  const bool full_tile = (row0 + M_TILE) <= n;

  const float eps = (epsp != nullptr) ? epsp[0] : 0.0f;
  const float oneeps = 1.0f + eps;

  // ---- stage 0: h tile -> LDS, float4 vectorized (rows clamped; EXEC stays full)
  {
    constexpr int DQ = DIN >> 2;
    for (int i = tid; i < M_TILE * DQ; i += 256) {
      int r = i / DQ, c = (i - r * DQ) << 2;
      int gr = row0 + r;
      if (gr > n - 1) gr = n - 1;
      long long off = (long long)gr * DIN + c;
      const float4 xv = *(const float4*)(X + off);
      const float4 av = *(const float4*)(AGG + off);
      float4 hv;
      hv.x = oneeps * xv.x + av.x;
      hv.y = oneeps * xv.y + av.y;
      hv.z = oneeps * xv.z + av.z;
      hv.w = oneeps * xv.w + av.w;
      *(float4*)(&sH[r][c]) = hv;
    }
  }
  __syncthreads();

  // WMMA f32 16x16x4 lane mapping:
  //   A: lanes 0-15 -> M=lane, K={0,1}; lanes 16-31 -> M=lane-16, K={2,3}
  //   B: lanes 0-15 -> N=lane, K={0,1}; lanes 16-31 -> N=lane-16, K={2,3}
  //   C/D: VGPR r -> row = r + 8*(lane>>4), col = lane&15
  const int m  = lane & 15;
  const int kh = (lane >> 4) << 1;    // 0 or 2

  // ---- stage 1: Y = relu(h @ W1 + b1)  (256 out cols -> 16 col-tiles, 8 per wave)
  {
    const int rsub = wave & 3;
    const float* arow = &sH[rsub * 16 + m][0];
    for (int j = 0; j < 8; ++j) {
      const int ct = (wave >> 2) + 2 * j;
      const int colg = ct * 16 + m;
      const float* wcol = WT1 + (size_t)colg * DIN;   // contiguous K
      v8f acc = {};
#pragma unroll 8
      for (int k = 0; k < DIN; k += 4) {
        v2f a = *(const v2f*)(arow + k + kh);
        v2f b = *(const v2f*)(wcol + k + kh);
        acc = __builtin_amdgcn_wmma_f32_16x16x4_f32(
            false, a, false, b, (short)0, acc, false, false);
      }
      const float bias = B1[colg];
#pragma unroll
      for (int r = 0; r < 8; ++r) {
        float v = acc[r] + bias;
        v = v > 0.0f ? v : 0.0f;               // ReLU
        sY[rsub * 16 + r + 8 * (lane >> 4)][colg] = v;
      }
    }
  }
  __syncthreads();

  // ---- stage 2: Z = Y @ W2 + b2 (+ BN + leaky relu)
  {
    const int rsub = wave & 3;
    constexpr int NCT = DOUT >> 4;
    const float* arow = &sY[rsub * 16 + m][0];
    for (int j = 0; j < (NCT >> 1); ++j) {
      const int ct = (wave >> 2) + 2 * j;
      const int colg = ct * 16 + m;
      const float* wcol = WT2 + (size_t)colg * H_DIM;
      v8f acc = {};
#pragma unroll 8
      for (int k = 0; k < H_DIM; k += 4) {
        v2f a = *(const v2f*)(arow + k + kh);
        v2f b = *(const v2f*)(wcol + k + kh);
        acc = __builtin_amdgcn_wmma_f32_16x16x4_f32(
            false, a, false, b, (short)0, acc, false, false);
      }
      float bias = B2[colg];
      float scale = 1.0f, shift = 0.0f;
      if (DO_BN) {
        scale = bn_gamma[colg] * rsqrtf(bn_var[colg] + 1e-5f);
        shift = bn_beta[colg] - bn_mean[colg] * scale;
      }
      float vr[8];
#pragma unroll
      for (int r = 0; r < 8; ++r) {
        float v = acc[r] + bias;
        if (DO_BN) {
          v = v * scale + shift;
          v = v >= 0.0f ? v : 0.01f * v;       // leaky relu
        }
        vr[r] = v;
      }
      const int rbase = row0 + rsub * 16 + 8 * (lane >> 4);
      if (full_tile) {                          // uniform fast path (781/782 blocks)
#pragma unroll
        for (int r = 0; r < 8; ++r)
          OUT[(long long)(rbase + r) * DOUT + colg] = vr[r];
      } else {
#pragma unroll
        for (int r = 0; r < 8; ++r)
          if (rbase + r < n)
            OUT[(long long)(rbase + r) * DOUT + colg] = vr[r];
      }
    }
  }
}

// ---------------------------------------------------------------- launcher
extern "C" void kernel_launch(void* const* d_in, const int* in_sizes, int n_in,
                              void* d_out, int out_size, void* d_ws, size_t ws_size,
                              hipStream_t stream) {
  const float* x     = (const float*)d_in[0];
  const int*   ei    = (const int*)  d_in[1];
  const float* eps0  = (const float*)d_in[2];
  const float* W01   = (const float*)d_in[3];
  const float* b01   = (const float*)d_in[4];
  const float* W02   = (const float*)d_in[5];
  const float* b02   = (const float*)d_in[6];
  const float* W11   = (const float*)d_in[7];
  const float* b11   = (const float*)d_in[8];
  const float* W12   = (const float*)d_in[9];
  const float* b12   = (const float*)d_in[10];
  const float* eps2  = (const float*)d_in[11];
  const float* W21   = (const float*)d_in[12];
  const float* b21   = (const float*)d_in[13];
  const float* W22   = (const float*)d_in[14];
  const float* b22   = (const float*)d_in[15];
  const float* gam   = (const float*)d_in[16];
  const float* bet   = (const float*)d_in[17];
  const float* mu    = (const float*)d_in[18];
  const float* var   = (const float*)d_in[19];

  const int N = in_sizes[0] / IN_DIM;
  const int E = in_sizes[1] / 2;
  const int* src = ei;
  const int* dst = ei + E;

  float* h1  = (float*)d_ws;                       // [N, 256]
  float* h2  = h1 + (size_t)N * H_DIM;             // [N, 256]
  float* agg = h2 + (size_t)N * H_DIM;             // [N, 256]
  float* wt1 = agg + (size_t)N * H_DIM;            // [256 * 256] transposed W1 of current layer
  float* wt2 = wt1 + (size_t)H_DIM * H_DIM;        // [256 * 256] transposed W2 of current layer

  const int mlp_blocks = (N + M_TILE - 1) / M_TILE;
  float* outp = (float*)d_out;

  // ---------- layer 0: din=128 -> H, BN+lrelu ----------
  {
    transpose_f32<<<(IN_DIM * H_DIM + 255) / 256, 256, 0, stream>>>(W01, wt1, IN_DIM, H_DIM);
    transpose_f32<<<(H_DIM * H_DIM + 255) / 256, 256, 0, stream>>>(W02, wt2, H_DIM, H_DIM);
    long long cnt = (long long)N * IN_DIM;
    zero_f32<<<(int)((cnt + 255) / 256), 256, 0, stream>>>(agg, cnt);
    long long work = (long long)E * (IN_DIM / 4);
    scatter_add_f32<<<(int)((work + 255) / 256), 256, 0, stream>>>(x, src, dst, agg, E, IN_DIM);
    gin_mlp<IN_DIM, H_DIM, 1><<<mlp_blocks, 256, 0, stream>>>(
        x, agg, eps0, wt1, b01, wt2, b02, gam, bet, mu, var, h1, N);
  }
  // ---------- layer 1: din=256 -> H, eps=0, BN+lrelu ----------
  {
    transpose_f32<<<(H_DIM * H_DIM + 255) / 256, 256, 0, stream>>>(W11, wt1, H_DIM, H_DIM);
    transpose_f32<<<(H_DIM * H_DIM + 255) / 256, 256, 0, stream>>>(W12, wt2, H_DIM, H_DIM);
    long long cnt = (long long)N * H_DIM;
    zero_f32<<<(int)((cnt + 255) / 256), 256, 0, stream>>>(agg, cnt);
    long long work = (long long)E * (H_DIM / 4);
    scatter_add_f32<<<(int)((work + 255) / 256), 256, 0, stream>>>(h1, src, dst, agg, E, H_DIM);
    gin_mlp<H_DIM, H_DIM, 1><<<mlp_blocks, 256, 0, stream>>>(
        h1, agg, nullptr, wt1, b11, wt2, b12, gam, bet, mu, var, h2, N);
  }
  // ---------- layer 2: din=256 -> OUT=64, no BN ----------
  {
    transpose_f32<<<(H_DIM * H_DIM + 255) / 256, 256, 0, stream>>>(W21, wt1, H_DIM, H_DIM);
    transpose_f32<<<(H_DIM * OUT_DIM + 255) / 256, 256, 0, stream>>>(W22, wt2, H_DIM, OUT_DIM);
    long long cnt = (long long)N * H_DIM;
    zero_f32<<<(int)((cnt + 255) / 256), 256, 0, stream>>>(agg, cnt);
    long long work = (long long)E * (H_DIM / 4);
    scatter_add_f32<<<(int)((work + 255) / 256), 256, 0, stream>>>(h2, src, dst, agg, E, H_DIM);
    gin_mlp<H_DIM, OUT_DIM, 0><<<mlp_blocks, 256, 0, stream>>>(
        h2, agg, eps2, wt1, b21, wt2, b22,
        nullptr, nullptr, nullptr, nullptr, outp, N);
  }
}